// ProductSpaceMessagePassing_30124900614678
// MI455X (gfx1250) — compile-verified
//
#include <hip/hip_runtime.h>

#define NN 50000
#define NE 800000
#define DD 128
#define NL 2
#define EPSF 1e-7f

typedef __attribute__((ext_vector_type(2))) float v2f;
typedef __attribute__((ext_vector_type(8))) float v8f;

static __device__ __forceinline__ float wave_sum(float v) {
#pragma unroll
  for (int off = 16; off > 0; off >>= 1) v += __shfl_xor(v, off, 32);
  return v;
}

__global__ void fill_zero_kernel(float* __restrict__ p, long n) {
  long i = (long)blockIdx.x * blockDim.x + threadIdx.x;
  long st = (long)gridDim.x * blockDim.x;
  for (; i < n; i += st) p[i] = 0.f;
}

__global__ void degree_kernel(const int* __restrict__ src, const int* __restrict__ dst,
                              float* __restrict__ deg_out, float* __restrict__ deg_in,
                              int e_cnt) {
  int e = blockIdx.x * blockDim.x + threadIdx.x;
  if (e < e_cnt) {
    unsafeAtomicAdd(&deg_out[src[e]], 1.f);
    unsafeAtomicAdd(&deg_in[dst[e]], 1.f);
  }
}

__global__ void norm_kernel(const float* __restrict__ deg_out, const float* __restrict__ deg_in,
                            float* __restrict__ norm_out, float* __restrict__ norm_in,
                            float* __restrict__ inv_in, int n) {
  int i = blockIdx.x * blockDim.x + threadIdx.x;
  if (i < n) {
    float o = deg_out[i], d = deg_in[i];
    norm_out[i] = (o > 0.f) ? rsqrtf(o) : 0.f;
    norm_in[i]  = (d > 0.f) ? rsqrtf(d) : 0.f;
    inv_in[i]   = (d > 0.f) ? (1.f / d) : 0.f;
  }
}

// y[row][:] = x[row][:] * s[row]   (one float4 per thread)
__global__ void scale_rows_kernel(const float* __restrict__ x, const float* __restrict__ s,
                                  float* __restrict__ y, int n_rows) {
  int t = blockIdx.x * blockDim.x + threadIdx.x;
  int total = n_rows * (DD / 4);
  if (t >= total) return;
  int row = t / (DD / 4);
  float4 v = ((const float4*)x)[t];
  float sc = s[row];
  v.x *= sc; v.y *= sc; v.z *= sc; v.w *= sc;
  ((float4*)y)[t] = v;
}

// one wave32 per edge; each lane handles 4 consecutive features (coalesced 512B row read)
__global__ void scatter_add_kernel(const float* __restrict__ x, const int* __restrict__ src,
                                   const int* __restrict__ dst, float* __restrict__ agg,
                                   int e_cnt) {
  long t = (long)blockIdx.x * blockDim.x + threadIdx.x;
  if (t >= (long)e_cnt * 32) return;
  int e = (int)(t >> 5);
  int q = ((int)t & 31) << 2;
  int s = src[e], d = dst[e];
  float4 v = *(const float4*)(x + (long)s * DD + q);
  float* a = agg + (long)d * DD + q;
  unsafeAtomicAdd(a + 0, v.x);
  unsafeAtomicAdd(a + 1, v.y);
  unsafeAtomicAdd(a + 2, v.z);
  unsafeAtomicAdd(a + 3, v.w);
}

// Poincare logmap0: artanh(clip(||y||)) * y / max(||y||, eps)
__global__ void logmap0_kernel(const float* __restrict__ x, float* __restrict__ y, int n_rows) {
  int gt = blockIdx.x * blockDim.x + threadIdx.x;
  int row = gt >> 5, lane = gt & 31;
  if (row >= n_rows) return;
  float4 v = *(const float4*)(x + (long)row * DD + lane * 4);
  float nrm = sqrtf(wave_sum(v.x * v.x + v.y * v.y + v.z * v.z + v.w * v.w));
  float nc = fminf(fmaxf(nrm, EPSF), 1.f - 1e-5f);
  float sc = atanhf(nc) / fmaxf(nrm, EPSF);
  v.x *= sc; v.y *= sc; v.z *= sc; v.w *= sc;
  *(float4*)(y + (long)row * DD + lane * 4) = v;
}

// expmap0(x * pre[row]): tanh(||u||) * u / max(||u||, eps)
__global__ void expmap0_kernel(const float* __restrict__ x, const float* __restrict__ pre,
                               float* __restrict__ y, int n_rows) {
  int gt = blockIdx.x * blockDim.x + threadIdx.x;
  int row = gt >> 5, lane = gt & 31;
  if (row >= n_rows) return;
  float p = pre[row];
  float4 v = *(const float4*)(x + (long)row * DD + lane * 4);
  v.x *= p; v.y *= p; v.z *= p; v.w *= p;
  float nrm = sqrtf(wave_sum(v.x * v.x + v.y * v.y + v.z * v.z + v.w * v.w));
  float sc = tanhf(nrm) / fmaxf(nrm, EPSF);
  v.x *= sc; v.y *= sc; v.z *= sc; v.w *= sc;
  *(float4*)(y + (long)row * DD + lane * 4) = v;
}

// l2norm(x * (pre ? pre[row] : 1))
__global__ void l2norm_kernel(const float* __restrict__ x, const float* __restrict__ pre,
                              float* __restrict__ y, int n_rows) {
  int gt = blockIdx.x * blockDim.x + threadIdx.x;
  int row = gt >> 5, lane = gt & 31;
  if (row >= n_rows) return;
  float p = pre ? pre[row] : 1.f;
  float4 v = *(const float4*)(x + (long)row * DD + lane * 4);
  v.x *= p; v.y *= p; v.z *= p; v.w *= p;
  float nrm = sqrtf(wave_sum(v.x * v.x + v.y * v.y + v.z * v.z + v.w * v.w));
  float sc = 1.f / fmaxf(nrm, 1e-12f);
  v.x *= sc; v.y *= sc; v.z *= sc; v.w *= sc;
  *(float4*)(y + (long)row * DD + lane * 4) = v;
}

// Y = act( (X * row_scale) @ W^T + bias ) using V_WMMA_F32_16X16X4_F32.
// Block = 256 threads (8 waves); block b owns rows [16b,16b+16); wave w owns cols [16w,16w+16).
__global__ __launch_bounds__(256) void gemm_nt_wmma(const float* __restrict__ X,
                                                    const float* __restrict__ W,
                                                    const float* __restrict__ bias,
                                                    const float* __restrict__ row_scale,
                                                    float* __restrict__ Y, int act) {
  __shared__ float As[16][132];  // stride 132 -> conflict-free column reads (64 banks)
  const int row0 = blockIdx.x * 16;
  const int tid = threadIdx.x;
#pragma unroll
  for (int i = 0; i < 8; ++i) {
    int idx = tid + i * 256;          // 16*128 = 2048 elements, coalesced
    int r = idx >> 7, c = idx & 127;
    float v = X[(long)(row0 + r) * DD + c];
    if (row_scale) v *= row_scale[row0 + r];
    As[r][c] = v;
  }
  __syncthreads();

  const int lane = tid & 31;
  const int colt = tid >> 5;           // column tile 0..7
  const int lmod = lane & 15;
  const int lhi  = lane >> 4;          // 0/1 selects K pair {0,1} vs {2,3}
  // A 16x4 fp32 layout: lane<16 -> (M=lane, K=k0+{0,1}); lane>=16 -> (M=lane-16, K=k0+{2,3})
  const float* Ap = &As[lmod][2 * lhi];
  // B 4x16 fp32 layout (B[k][n] = W[n][k]): lane<16 -> rows K=k0+{0,1} at N=lane, etc.
  const float* Wp = W + (long)(colt * 16 + lmod) * DD + 2 * lhi;

  v8f acc = {};
#pragma unroll
  for (int k0 = 0; k0 < DD; k0 += 4) {
    v2f a = *(const v2f*)(Ap + k0);
    v2f b = *(const v2f*)(Wp + k0);
    acc = __builtin_amdgcn_wmma_f32_16x16x4_f32(false, a, false, b, (short)0, acc,
                                                false, false);
  }

  // C/D layout: VGPR v holds row M = v + 8*lhi, col N = lmod
  float bc = bias[colt * 16 + lmod];
#pragma unroll
  for (int vi = 0; vi < 8; ++vi) {
    int r = row0 + vi + 8 * lhi;
    float val = acc[vi] + bc;
    if (act) val = (val > 0.f) ? val : 0.2f * val;
    Y[(long)r * DD + colt * 16 + lmod] = val;
  }
}

extern "C" void kernel_launch(void* const* d_in, const int* in_sizes, int n_in,
                              void* d_out, int out_size, void* d_ws, size_t ws_size,
                              hipStream_t stream) {
  (void)in_sizes; (void)n_in; (void)out_size; (void)ws_size;
  const float* e_emb = (const float*)d_in[0];
  const float* b_emb = (const float*)d_in[1];
  const float* s_emb = (const float*)d_in[2];
  const float* eW = (const float*)d_in[3];
  const float* eB = (const float*)d_in[4];
  const float* bW = (const float*)d_in[5];
  const float* bB = (const float*)d_in[6];
  const float* sW = (const float*)d_in[7];
  const float* sB = (const float*)d_in[8];
  const int* src = (const int*)d_in[9];
  const int* dst = (const int*)d_in[10];
  float* out = (float*)d_out;

  const long ND = (long)NN * DD;
  float* ws = (float*)d_ws;
  float* X    = ws;
  float* T    = ws + ND;
  float* Agg  = ws + 2 * ND;
  float* e1   = ws + 3 * ND;
  float* b1   = ws + 4 * ND;
  float* s1   = ws + 5 * ND;
  float* dego = ws + 6 * ND;          // N
  float* degi = dego + NN;            // N
  float* nout = degi + NN;            // N
  float* nin  = nout + NN;            // N
  float* iin  = nin + NN;             // N

  const int rowsPerD4 = NN * (DD / 4);                 // float4 elementwise threads
  const int gScale   = (rowsPerD4 + 255) / 256;        // 6250
  const int gRowWave = ((NN * 32) + 255) / 256;        // 6250 (wave per row)
  const long scatterT = (long)NE * 32;
  const int gScatter = (int)((scatterT + 255) / 256);  // 100000
  const int gGemm    = NN / 16;                        // 3125 (exact)

  // degrees + norms
  fill_zero_kernel<<<512, 256, 0, stream>>>(dego, 2L * NN);
  degree_kernel<<<(NE + 255) / 256, 256, 0, stream>>>(src, dst, dego, degi, NE);
  norm_kernel<<<(NN + 255) / 256, 256, 0, stream>>>(dego, degi, nout, nin, iin, NN);

  const float* e_cur = e_emb;
  const float* b_cur = b_emb;
  const float* s_cur = s_emb;

  for (int l = 0; l < NL; ++l) {
    float* e_nxt = (l == NL - 1) ? out            : e1;
    float* b_nxt = (l == NL - 1) ? out + ND       : b1;
    float* s_nxt = (l == NL - 1) ? out + 2 * ND   : s1;
    const float* eWl = eW + (long)l * DD * DD;
    const float* bWl = bW + (long)l * DD * DD;
    const float* sWl = sW + (long)l * DD * DD;

    // ---- Euclidean: agg(e * nout) * nin -> GEMM(+bias, leaky) ----
    scale_rows_kernel<<<gScale, 256, 0, stream>>>(e_cur, nout, X, NN);
    fill_zero_kernel<<<4096, 256, 0, stream>>>(Agg, ND);
    scatter_add_kernel<<<gScatter, 256, 0, stream>>>(X, src, dst, Agg, NE);
    gemm_nt_wmma<<<gGemm, 256, 0, stream>>>(Agg, eWl, eB + (long)l * DD, nin, e_nxt, 1);

    // ---- Hyperbolic: logmap0 -> GEMM -> agg*inv_in -> expmap0 ----
    logmap0_kernel<<<gRowWave, 256, 0, stream>>>(b_cur, X, NN);
    gemm_nt_wmma<<<gGemm, 256, 0, stream>>>(X, bWl, bB + (long)l * DD, nullptr, T, 0);
    fill_zero_kernel<<<4096, 256, 0, stream>>>(Agg, ND);
    scatter_add_kernel<<<gScatter, 256, 0, stream>>>(T, src, dst, Agg, NE);
    expmap0_kernel<<<gRowWave, 256, 0, stream>>>(Agg, iin, b_nxt, NN);

    // ---- Spherical: l2norm -> GEMM -> agg*inv_in -> l2norm ----
    l2norm_kernel<<<gRowWave, 256, 0, stream>>>(s_cur, nullptr, X, NN);
    gemm_nt_wmma<<<gGemm, 256, 0, stream>>>(X, sWl, sB + (long)l * DD, nullptr, T, 0);
    fill_zero_kernel<<<4096, 256, 0, stream>>>(Agg, ND);
    scatter_add_kernel<<<gScatter, 256, 0, stream>>>(T, src, dst, Agg, NE);
    l2norm_kernel<<<gRowWave, 256, 0, stream>>>(Agg, iin, s_nxt, NN);

    e_cur = e_nxt; b_cur = b_nxt; s_cur = s_nxt;
  }
}